// SmoothLDDTLoss_53300544143993
// MI455X (gfx1250) — compile-verified
//
#include <hip/hip_runtime.h>

typedef __attribute__((ext_vector_type(2))) float v2f;
typedef __attribute__((ext_vector_type(8))) float v8f;

#define NPTS   4096
#define BATCH  2
#define TILES  (NPTS / 16)              // 256 tile rows/cols
#define TRI    (TILES * (TILES + 1) / 2) // 32896 upper-tri tiles per batch
#define WPB    8                         // waves per block
#define BLOCK  (WPB * 32)

__global__ void lddt_init_kernel(double* acc) { acc[0] = 0.0; }

__global__ __launch_bounds__(BLOCK)
void lddt_tile_kernel(const float* __restrict__ x,
                      const float* __restrict__ xg,
                      double* __restrict__ acc)
{
    const int lane = threadIdx.x & 31;
    const int wave = threadIdx.x >> 5;
    const int idx  = lane & 15;   // column id (and row-within-group for A)
    const int half = lane >> 4;   // A/B: K={0,1} vs K={2,3}; C/D: rows 0-7 vs 8-15

    // ---- triangular tile index -> (batch, tl, tm) with tl <= tm ----
    const int tile = blockIdx.x * WPB + wave;      // < BATCH*TRI = 65792
    const int b = tile / TRI;
    const int t = tile % TRI;
    // off(tl) = tl*TILES - tl*(tl-1)/2 ; seed with sqrt, fix up exactly.
    const float twoT1 = 2.0f * TILES + 1.0f;
    int tl = (int)((twoT1 - sqrtf(twoT1 * twoT1 - 8.0f * (float)t)) * 0.5f);
    if (tl < 0) tl = 0;
    if (tl > TILES - 1) tl = TILES - 1;
    while (tl > 0 && t < tl * TILES - tl * (tl - 1) / 2) --tl;
    while (t >= (tl + 1) * TILES - (tl + 1) * tl / 2) ++tl;
    const int tm = tl + (t - (tl * TILES - tl * (tl - 1) / 2));
    const int l0 = tl * 16, m0 = tm * 16;

    const float* xb = x  + (size_t)b * NPTS * 3;
    const float* gb = xg + (size_t)b * NPTS * 3;

    const int li = (l0 + idx) * 3;
    const int mi = (m0 + idx) * 3;
    const float xl0 = xb[li + 0], xl1 = xb[li + 1], xl2 = xb[li + 2];
    const float gl0 = gb[li + 0], gl1 = gb[li + 1], gl2 = gb[li + 2];
    const float xm0 = xb[mi + 0], xm1 = xb[mi + 1], xm2 = xb[mi + 2];
    const float gm0 = gb[mi + 0], gm1 = gb[mi + 1], gm2 = gb[mi + 2];

    const float nl  = xl0*xl0 + xl1*xl1 + xl2*xl2;   // row norm -> K=3 slot of A
    const float nlg = gl0*gl0 + gl1*gl1 + gl2*gl2;
    const float nm  = xm0*xm0 + xm1*xm1 + xm2*xm2;   // col norm: lane-local add
    const float nmg = gm0*gm0 + gm1*gm1 + gm2*gm2;

    // A[l] = ( x, y, z, ||x_l||^2 ) ; lanes 0-15 hold K=0,1 ; lanes 16-31 K=2,3.
    v2f a, ag, bm, bg;
    a.x  = half ? xl2 : xl0;   a.y  = half ? nl  : xl1;
    ag.x = half ? gl2 : gl0;   ag.y = half ? nlg : gl1;
    // B[:,m] = ( -2x, -2y, -2z, 1 ) ; same K striping across lane halves.
    bm.x = half ? -2.0f*xm2 : -2.0f*xm0;   bm.y = half ? 1.0f : -2.0f*xm1;
    bg.x = half ? -2.0f*gm2 : -2.0f*gm0;   bg.y = half ? 1.0f : -2.0f*gm1;

    v8f zero = {};
    // D[l][m] = ||x_l||^2 - 2 x_l.x_m   (distance^2 minus lane-local col norm)
    v8f dotx = __builtin_amdgcn_wmma_f32_16x16x4_f32(
        false, a,  false, bm, (short)0, zero, false, false);
    v8f dotg = __builtin_amdgcn_wmma_f32_16x16x4_f32(
        false, ag, false, bg, (short)0, zero, false, false);

    // sigmoid(tau - d) = 1 / (1 + e^d * e^-tau): 1 exp + 4 rcp per element.
    const float c05 = 0.60653065971263342360f;  // e^-0.5
    const float c10 = 0.36787944117144232160f;  // e^-1
    const float c20 = 0.13533528323661269189f;  // e^-2
    const float c40 = 0.01831563888873418029f;  // e^-4

    float sum = 0.0f;
    #pragma unroll
    for (int r = 0; r < 8; ++r) {
        const int row = r + 8 * half;            // C/D: VGPR r -> row r+8*half
        const float dx = fmaxf(dotx[r] + nm,  0.0f);
        const float dg = fmaxf(dotg[r] + nmg, 0.0f);
        const float delta = fabsf(dg - dx);
        const float e = __expf(delta);           // v_exp_f32; inf-safe (rcp->0)
        float eps = __builtin_amdgcn_rcpf(fmaf(e, c05, 1.0f))
                  + __builtin_amdgcn_rcpf(fmaf(e, c10, 1.0f))
                  + __builtin_amdgcn_rcpf(fmaf(e, c20, 1.0f))
                  + __builtin_amdgcn_rcpf(fmaf(e, c40, 1.0f));
        // mask: c_lm contains (nuc|~nuc) == true; only the diagonal is excluded.
        const bool offdiag = (l0 + row) != (m0 + idx);
        sum += offdiag ? eps : 0.0f;
    }
    // upper-tri tile counts for both (l,m) and (m,l); diagonal tile counts once.
    sum *= 0.25f * ((tl == tm) ? 1.0f : 2.0f);

    #pragma unroll
    for (int off = 16; off > 0; off >>= 1)
        sum += __shfl_down(sum, off, 32);

    __shared__ float wsum[WPB];
    if (lane == 0) wsum[wave] = sum;
    __syncthreads();
    if (threadIdx.x == 0) {
        float s = 0.0f;
        #pragma unroll
        for (int w = 0; w < WPB; ++w) s += wsum[w];
        atomicAdd(acc, (double)s);               // global_atomic_add_f64
    }
}

__global__ void lddt_finalize_kernel(const double* __restrict__ acc,
                                     float* __restrict__ out)
{
    const double denom = (double)BATCH * (double)NPTS * (double)(NPTS - 1);
    out[0] = (float)(1.0 - acc[0] / denom);
}

extern "C" void kernel_launch(void* const* d_in, const int* in_sizes, int n_in,
                              void* d_out, int out_size, void* d_ws, size_t ws_size,
                              hipStream_t stream) {
    (void)in_sizes; (void)n_in; (void)out_size; (void)ws_size;
    const float* x  = (const float*)d_in[0];   // x_l   [B, N, 3] f32
    const float* xg = (const float*)d_in[1];   // xGT_l [B, N, 3] f32
    // d_in[2], d_in[3] (is_dna, is_rna) are mathematically unused:
    // the reference mask contains (nuc | ~nuc) and is always true off-diagonal.
    float*  out = (float*)d_out;
    double* acc = (double*)d_ws;

    lddt_init_kernel<<<1, 1, 0, stream>>>(acc);

    const int total_tiles = BATCH * TRI;             // 65792 (upper triangle)
    const int blocks = (total_tiles + WPB - 1) / WPB; // 8224 blocks of 8 waves
    lddt_tile_kernel<<<blocks, BLOCK, 0, stream>>>(x, xg, acc);

    lddt_finalize_kernel<<<1, 1, 0, stream>>>(acc, out);
}